// QuantumLayer_83416854823484
// MI455X (gfx1250) — compile-verified
//
#include <hip/hip_runtime.h>

typedef __attribute__((ext_vector_type(2))) float v2f;
typedef __attribute__((ext_vector_type(8))) float v8f;

#define NQ  12
#define NL  4
#define DIM 4096   // 2^12 amplitudes
#define BLK 256    // 8 wave32

// State viewed as psi[i0][i1][i2], each index 4 bits; qubit w <-> bit (11-w).
// Pass P multiplies a 16x16 matrix along index iP. Tile t / element (k,n):
//   P0: psi[i0=k, i1=t, i2=n]   P1: psi[t, k, n]   P2: psi[t, n, k]
__device__ __forceinline__ int sidx(int P, int t, int k, int n) {
    if (P == 0) return (k << 8) | (t << 4) | n;
    if (P == 1) return (t << 8) | (k << 4) | n;
    return (t << 8) | (n << 4) | k;
}

// The CNOT ring CNOT(i,(i+1)%12), applied in order, as a GF(2)-linear
// permutation of basis indices (target_bit ^= control_bit).
__device__ __forceinline__ int cnot_ring(int j) {
#pragma unroll
    for (int i = 0; i < NQ; ++i) {
        int pc = 11 - i;
        int pt = 11 - ((i + 1) % NQ);
        j ^= ((j >> pc) & 1) << pt;
    }
    return j;
}

__global__ void __launch_bounds__(BLK)
qsim12(const float* __restrict__ x, const float* __restrict__ wgt,
       float* __restrict__ out)
{
    __shared__ float2 S[DIM];            // interleaved (re, im) state  (32 KB)
    __shared__ float2 K[NL * 3 * 256];   // 12 complex 16x16 gate matrices (24 KB)

    const int tid  = threadIdx.x;
    const int b    = blockIdx.x;
    const int lane = tid & 31;
    const int wv   = tid >> 5;   // wave 0..7
    const int lg   = lane >> 4;  // half-wave 0/1
    const int ln   = lane & 15;

    // ---- initial state: RY(x)|0...0> is a product state ----
    // j = e*256 + tid : amplitude = A_hi[e] (qubits 0..3) * B_lo[tid] (qubits 4..11)
    float ch[NQ], sh[NQ];
#pragma unroll
    for (int w = 0; w < NQ; ++w)
        __sincosf(0.5f * x[b * NQ + w], &sh[w], &ch[w]);   // hw trans: args are O(1)

    float blo = 1.0f;
#pragma unroll
    for (int w = 4; w < NQ; ++w)                 // qubit w <-> bit (11-w) of tid
        blo *= ((tid >> (11 - w)) & 1) ? sh[w] : ch[w];
#pragma unroll
    for (int e = 0; e < DIM / BLK; ++e) {
        float ahi = 1.0f;
#pragma unroll
        for (int w = 0; w < 4; ++w)              // qubit w <-> bit (3-w) of e
            ahi *= ((e >> (3 - w)) & 1) ? sh[w] : ch[w];
        S[tid + BLK * e] = make_float2(ahi * blo, 0.0f);
    }

    // ---- build all 12 K(l,g) = kron of RX gates, once ----
    // Entry (r,c): (-i)^popc(r^c) * prod_u (diff_u ? sin : cos)(w/2).
    {
        int r = tid >> 4, c = tid & 15;
        int dxo = r ^ c;
        int t4 = __popc(dxo) & 3;
#pragma unroll
        for (int m = 0; m < NL * 3; ++m) {       // m = l*3 + g
            float p = 1.0f;
#pragma unroll
            for (int u = 0; u < 4; ++u) {
                float s, cth;
                __sincosf(0.5f * wgt[(m / 3) * NQ + (m % 3) * 4 + u], &s, &cth);
                p *= ((dxo >> (3 - u)) & 1) ? s : cth;
            }
            float kr = (t4 == 0) ? p : ((t4 == 2) ? -p : 0.0f);
            float ki = (t4 == 1) ? -p : ((t4 == 3) ? p : 0.0f);
            K[m * 256 + tid] = make_float2(kr, ki);
        }
    }
    __syncthreads();

    for (int l = 0; l < NL; ++l) {
#pragma unroll
        for (int g = 0; g < 3; ++g) {
            const float2* Km = &K[(l * 3 + g) * 256];

            // A fragments (16x4 f32, v2f/lane): element = K[m=ln][k0+j]
            v2f ar[4], ai[4], an[4];
#pragma unroll
            for (int c4 = 0; c4 < 4; ++c4) {
                int base = ln * 16 + 4 * c4 + 2 * lg;
                float2 k0 = Km[base], k1 = Km[base + 1];    // b64 loads
                ar[c4].x = k0.x;   ar[c4].y = k1.x;
                ai[c4].x = k0.y;   ai[c4].y = k1.y;
                an[c4].x = -k0.y;  an[c4].y = -k1.y;        // -Ki (f32 WMMA has no A-neg)
            }

#pragma unroll
            for (int tt = 0; tt < 2; ++tt) {     // 16 tiles / 8 waves
                const int t = wv + 8 * tt;
                // B fragments (4x16 f32): element = X[k0+j][n=ln]; one b64 per element
                v2f br[4], bi[4];
#pragma unroll
                for (int c4 = 0; c4 < 4; ++c4) {
                    int k0 = 4 * c4 + 2 * lg;
                    float2 f0 = S[sidx(g, t, k0,     ln)];
                    float2 f1 = S[sidx(g, t, k0 + 1, ln)];
                    br[c4].x = f0.x;  br[c4].y = f1.x;
                    bi[c4].x = f0.y;  bi[c4].y = f1.y;
                }
                // complex 16x16x16: Dr = Kr*Xr - Ki*Xi ; Di = Kr*Xi + Ki*Xr
                v8f dr = {}, di = {};
#pragma unroll
                for (int c4 = 0; c4 < 4; ++c4)
                    dr = __builtin_amdgcn_wmma_f32_16x16x4_f32(false, ar[c4], false, br[c4], (short)0, dr, false, false);
#pragma unroll
                for (int c4 = 0; c4 < 4; ++c4)
                    dr = __builtin_amdgcn_wmma_f32_16x16x4_f32(false, an[c4], false, bi[c4], (short)0, dr, false, false);
#pragma unroll
                for (int c4 = 0; c4 < 4; ++c4)
                    di = __builtin_amdgcn_wmma_f32_16x16x4_f32(false, ar[c4], false, bi[c4], (short)0, di, false, false);
#pragma unroll
                for (int c4 = 0; c4 < 4; ++c4)
                    di = __builtin_amdgcn_wmma_f32_16x16x4_f32(false, ai[c4], false, br[c4], (short)0, di, false, false);
                // D layout: vgpr i, lane (lg,ln) -> row i+8*lg, col ln; one b64 per element
#pragma unroll
                for (int i = 0; i < 8; ++i)
                    S[sidx(g, t, i + 8 * lg, ln)] = make_float2(dr[i], di[i]);
            }
            __syncthreads();
        }

        // ---- CNOT ring: fixed basis permutation, in place over a barrier ----
        float2 pv[DIM / BLK];
#pragma unroll
        for (int e = 0; e < DIM / BLK; ++e)
            pv[e] = S[tid + BLK * e];
        __syncthreads();
#pragma unroll
        for (int e = 0; e < DIM / BLK; ++e)
            S[cnot_ring(tid + BLK * e)] = pv[e];
        __syncthreads();
    }

    // ---- <Z_w> = sum_j (-1)^{bit_{11-w}(j)} |psi_j|^2 ----
    float acc[NQ];
#pragma unroll
    for (int w = 0; w < NQ; ++w) acc[w] = 0.0f;
#pragma unroll
    for (int e = 0; e < DIM / BLK; ++e) {
        int j = tid + BLK * e;
        float2 a = S[j];
        float p = a.x * a.x + a.y * a.y;
#pragma unroll
        for (int w = 0; w < NQ; ++w)
            acc[w] += ((j >> (11 - w)) & 1) ? -p : p;
    }
    __syncthreads();              // state no longer needed; reuse S as scratch
    float* red = (float*)S;       // 12*256 = 3072 floats << 8192 available
#pragma unroll
    for (int w = 0; w < NQ; ++w) red[w * 256 + tid] = acc[w];
    __syncthreads();
    for (int off = 128; off >= 1; off >>= 1) {
        if (tid < off) {
#pragma unroll
            for (int w = 0; w < NQ; ++w)
                red[w * 256 + tid] += red[w * 256 + tid + off];
        }
        __syncthreads();
    }
    if (tid < NQ) out[b * NQ + tid] = red[tid * 256];
}

extern "C" void kernel_launch(void* const* d_in, const int* in_sizes, int n_in,
                              void* d_out, int out_size, void* d_ws, size_t ws_size,
                              hipStream_t stream)
{
    (void)in_sizes; (void)n_in; (void)d_ws; (void)ws_size; (void)out_size;
    const float* x   = (const float*)d_in[0];   // (4096, 12) f32
    const float* wgt = (const float*)d_in[1];   // (4, 12) f32
    float*       out = (float*)d_out;           // (4096, 12) f32
    qsim12<<<dim3(4096), dim3(BLK), 0, stream>>>(x, wgt, out);
}